// StressPredictorFromFp_78967268704762
// MI455X (gfx1250) — compile-verified
//
#include <hip/hip_runtime.h>

typedef __attribute__((ext_vector_type(2))) float v2f;
typedef __attribute__((ext_vector_type(8))) float v8f;

#define HIDN 128
#define PADK 132              // padded k-stride (dwords): bank stride 4 -> conflict-free
#define WAVES 8
#define ROWS_PER_WAVE 32      // M=32: two 16-row tiles share every B-operand load
#define ROWS_PER_BLOCK 256    // 8 waves * 32 rows

__global__ __launch_bounds__(256)
void stress_predictor_kernel(const float* __restrict__ F,
                             const float* __restrict__ W1, const float* __restrict__ b1,
                             const float* __restrict__ W2, const float* __restrict__ b2,
                             const float* __restrict__ W3, const float* __restrict__ b3,
                             float* __restrict__ out, int n, int nchunks) {
    extern __shared__ float smem[];
    float* w1c   = smem;                   // [128][4]   : W1^T, k contiguous, row 3 zero-padded
    float* w3c   = w1c + HIDN * 4;         // [16][132]  : W3^T, cols 4..15 zero-padded
    float* w2c   = w3c + 16 * PADK;        // [128][132] : W2^T, k contiguous
    float* hbase = w2c + HIDN * PADK;      // 8 waves * [32][132] activation staging

    const int tid = threadIdx.x;

    // ---- stage weights into B-operand-friendly (n-major, k-contiguous) LDS layout ----
    for (int i = tid; i < HIDN * 4; i += blockDim.x) {
        int nn = i >> 2, kk = i & 3;
        w1c[i] = (kk < 3) ? W1[kk * HIDN + nn] : 0.0f;            // W1 is [3][128]
    }
    for (int i = tid; i < 16 * HIDN; i += blockDim.x) {
        int nn = i >> 7, kk = i & 127;
        w3c[nn * PADK + kk] = (nn < 4) ? W3[kk * 4 + nn] : 0.0f;  // W3 is [128][4]
    }
    for (int i = tid; i < HIDN * HIDN; i += blockDim.x) {
        int nn = i >> 7, kk = i & 127;
        w2c[nn * PADK + kk] = W2[kk * HIDN + nn];                 // W2 is [128][128]
    }
    __syncthreads();

    const int lane  = tid & 31;
    const int w     = tid >> 5;
    const int cn    = lane & 15;                 // column / row-within-tile index
    const int khalf = (lane < 16) ? 0 : 2;       // K split for A/B operands
    const int hoff  = (lane < 16) ? 0 : 8;       // M split within a C/D tile
    float* hb     = hbase + w * (ROWS_PER_WAVE * PADK);
    float* hrowA0 = hb + cn * PADK;              // A-operand row, tile u=0 (rows 0..15)
    float* hrowA1 = hb + (16 + cn) * PADK;       // A-operand row, tile u=1 (rows 16..31)
    float* hrowL  = hb + lane * PADK;            // this lane's own row (x write / y read)

    // per-lane biases (column-indexed, constant across the 8 rows of each D tile)
    float bias1[8], bias2[8];
#pragma unroll
    for (int j = 0; j < 8; ++j) {
        bias1[j] = b1[j * 16 + cn];
        bias2[j] = b2[j * 16 + cn];
    }
    const float bias3 = (cn < 4) ? b3[cn] : 0.0f;

    for (int chunk = blockIdx.x; chunk < nchunks; chunk += gridDim.x) {
        const int row = chunk * ROWS_PER_BLOCK + w * ROWS_PER_WAVE + lane;
        const bool rowvalid = (row < n);

        // ---- closed-form 2x2 polar decomposition + invariants (one row per lane) ----
        float fa = 1.0f, fb = 0.0f, fc = 0.0f, fd = 1.0f;
        if (rowvalid) {
            float4 fv = ((const float4*)F)[row];
            fa = fv.x; fb = fv.y; fc = fv.z; fd = fv.w;
        }
        float det = fa * fd - fb * fc;
        float sgn = (det >= 0.0f) ? 1.0f : -1.0f;
        // R = (F + sgn*cof(F)) / ||.||,  cof(F) = [[d,-c],[-b,a]]
        float m00 = fa + sgn * fd, m01 = fb - sgn * fc;
        float m10 = fc - sgn * fb, m11 = fd + sgn * fa;
        float h2  = m00 * m00 + m01 * m01;       // == hp^2 (det>=0) or hq^2 (det<0)
        float hin = rsqrtf(fmaxf(h2, 1e-30f));
        float r00 = m00 * hin, r01 = m01 * hin, r10 = m10 * hin, r11 = m11 * hin;
        float hnm = h2 * hin;                    // = max(hp,hq) = sum of singular values
        hrowL[0] = hnm - 2.0f;                                    // I1
        hrowL[1] = fa * fa + fb * fb + fc * fc + fd * fd - 1.0f;  // I2
        hrowL[2] = det - 1.0f;                                    // I3
        hrowL[3] = 0.0f;                                          // K pad

        // ---- layer 1: [32x3(+pad)] @ [3x128], one K-step, B shared by both M tiles ----
        v8f acc0[8], acc1[8];
        {
            v2f a0 = *(const v2f*)(hrowA0 + khalf);
            v2f a1 = *(const v2f*)(hrowA1 + khalf);
#pragma unroll
            for (int j = 0; j < 8; ++j) {
                v2f bw = *(const v2f*)(w1c + (j * 16 + cn) * 4 + khalf);
                v8f z = {0.f, 0.f, 0.f, 0.f, 0.f, 0.f, 0.f, 0.f};
                acc0[j] = __builtin_amdgcn_wmma_f32_16x16x4_f32(
                    false, a0, false, bw, (short)0, z, false, false);
                acc1[j] = __builtin_amdgcn_wmma_f32_16x16x4_f32(
                    false, a1, false, bw, (short)0, z, false, false);
            }
        }
#pragma unroll
        for (int j = 0; j < 8; ++j)
#pragma unroll
            for (int r = 0; r < 8; ++r) {
                hb[(r + hoff) * PADK + j * 16 + cn]        = fmaxf(acc0[j][r] + bias1[j], 0.0f);
                hb[(16 + r + hoff) * PADK + j * 16 + cn]   = fmaxf(acc1[j][r] + bias1[j], 0.0f);
            }

        // ---- layer 2: [32x128] @ [128x128]; each B load feeds 2 WMMAs ----
#pragma unroll
        for (int j = 0; j < 8; ++j) {
            v8f z = {0.f, 0.f, 0.f, 0.f, 0.f, 0.f, 0.f, 0.f};
            acc0[j] = z; acc1[j] = z;
        }
        for (int k0 = 0; k0 < HIDN; k0 += 4) {
            v2f a0 = *(const v2f*)(hrowA0 + k0 + khalf);
            v2f a1 = *(const v2f*)(hrowA1 + k0 + khalf);
#pragma unroll
            for (int j = 0; j < 8; ++j) {
                v2f bw = *(const v2f*)(w2c + (j * 16 + cn) * PADK + k0 + khalf);
                acc0[j] = __builtin_amdgcn_wmma_f32_16x16x4_f32(
                    false, a0, false, bw, (short)0, acc0[j], false, false);
                acc1[j] = __builtin_amdgcn_wmma_f32_16x16x4_f32(
                    false, a1, false, bw, (short)0, acc1[j], false, false);
            }
        }
#pragma unroll
        for (int j = 0; j < 8; ++j)
#pragma unroll
            for (int r = 0; r < 8; ++r) {
                hb[(r + hoff) * PADK + j * 16 + cn]        = fmaxf(acc0[j][r] + bias2[j], 0.0f);
                hb[(16 + r + hoff) * PADK + j * 16 + cn]   = fmaxf(acc1[j][r] + bias2[j], 0.0f);
            }

        // ---- layer 3: [32x128] @ [128x4(+pad to 16)] ----
        v8f acc3a = {0.f, 0.f, 0.f, 0.f, 0.f, 0.f, 0.f, 0.f};
        v8f acc3b = {0.f, 0.f, 0.f, 0.f, 0.f, 0.f, 0.f, 0.f};
        for (int k0 = 0; k0 < HIDN; k0 += 4) {
            v2f a0 = *(const v2f*)(hrowA0 + k0 + khalf);
            v2f a1 = *(const v2f*)(hrowA1 + k0 + khalf);
            v2f bw = *(const v2f*)(w3c + cn * PADK + k0 + khalf);
            acc3a = __builtin_amdgcn_wmma_f32_16x16x4_f32(
                false, a0, false, bw, (short)0, acc3a, false, false);
            acc3b = __builtin_amdgcn_wmma_f32_16x16x4_f32(
                false, a1, false, bw, (short)0, acc3b, false, false);
        }
        // scatter y (cols 0..3 only) back to per-row LDS slots
        if (cn < 4) {
#pragma unroll
            for (int r = 0; r < 8; ++r) {
                hb[(r + hoff) * PADK + cn]      = acc3a[r] + bias3;
                hb[(16 + r + hoff) * PADK + cn] = acc3b[r] + bias3;
            }
        }

        // ---- epilogue: S = sym(y); P = R S; cauchy = P F^T (one row per lane) ----
        if (rowvalid) {
            float y00 = hrowL[0], y01 = hrowL[1], y10 = hrowL[2], y11 = hrowL[3];
            float s01 = 0.5f * (y01 + y10);
            float p00 = r00 * y00 + r01 * s01;
            float p01 = r00 * s01 + r01 * y11;
            float p10 = r10 * y00 + r11 * s01;
            float p11 = r10 * s01 + r11 * y11;
            float4 o;
            o.x = p00 * fa + p01 * fb;
            o.y = p00 * fc + p01 * fd;
            o.z = p10 * fa + p11 * fb;
            o.w = p10 * fc + p11 * fd;
            ((float4*)out)[row] = o;
        }
    }
}

extern "C" void kernel_launch(void* const* d_in, const int* in_sizes, int n_in,
                              void* d_out, int out_size, void* d_ws, size_t ws_size,
                              hipStream_t stream) {
    (void)n_in; (void)d_ws; (void)ws_size; (void)out_size;
    const float* F  = (const float*)d_in[0];
    const float* W1 = (const float*)d_in[1];
    const float* b1 = (const float*)d_in[2];
    const float* W2 = (const float*)d_in[3];
    const float* b2 = (const float*)d_in[4];
    const float* W3 = (const float*)d_in[5];
    const float* b3 = (const float*)d_in[6];
    float* out = (float*)d_out;

    const int n = in_sizes[0] / 4;                       // number of 2x2 matrices
    const int nchunks = (n + ROWS_PER_BLOCK - 1) / ROWS_PER_BLOCK;
    const int grid = nchunks < 4096 ? nchunks : 4096;
    const size_t shmem =
        (size_t)(HIDN * 4 + 16 * PADK + HIDN * PADK + WAVES * ROWS_PER_WAVE * PADK) * sizeof(float);

    hipLaunchKernelGGL(stress_predictor_kernel, dim3(grid), dim3(256), shmem, stream,
                       F, W1, b1, W2, b2, W3, b3, out, n, nchunks);
}